// JPEG_20349555048677
// MI455X (gfx1250) — compile-verified
//
#include <hip/hip_runtime.h>
#include <hip/hip_bf16.h>
#include <math.h>

typedef __attribute__((ext_vector_type(2))) float v2f;
typedef __attribute__((ext_vector_type(8))) float v8f;

#define PI_16 0.19634954084936207f   /* pi/16 */
#define RSQ2  0.70710678118654752f   /* 1/sqrt(2) */

// Quant tables in [u][v] order (reference stores the .T of the classic tables;
// YTAB below is already transposed to match q[u,v] = f[u,v]/table[u,v]).
__constant__ float YTAB[64] = {
  16,12,14,14, 18, 24, 49, 72,
  11,12,13,17, 22, 35, 64, 92,
  10,14,16,22, 37, 55, 78, 95,
  16,19,24,29, 56, 64, 87, 98,
  24,26,40,51, 68, 81,103,112,
  40,58,57,87,109,104,121,100,
  51,60,69,80,103,113,120,103,
  61,55,56,62, 77, 92,101, 99 };

__constant__ float CTAB[64] = {
  17,18,24,47,99,99,99,99,
  18,21,26,66,99,99,99,99,
  24,26,56,99,99,99,99,99,
  47,66,99,99,99,99,99,99,
  99,99,99,99,99,99,99,99,
  99,99,99,99,99,99,99,99,
  99,99,99,99,99,99,99,99,
  99,99,99,99,99,99,99,99 };

// ---------------------------------------------------------------------------
// Stage 1: RGB -> YCbCr + 4:2:0 average pooling.
// Planes live in workspace as plane[b][ww][hh], hh contiguous (matches the
// hh-contiguous last dim of pixel_inp[b,c,ww,hh]), so every access is a
// coalesced float2.
// ---------------------------------------------------------------------------
__global__ void k_rgb2ycc(const float* __restrict__ pix, float* __restrict__ Y,
                          float* __restrict__ Cb, float* __restrict__ Cr)
{
    int g   = blockIdx.x * blockDim.x + threadIdx.x;   // one 2x2 pixel quad
    int b   = g >> 14;
    int rem = g & 16383;
    int wwc = rem >> 7;
    int hhc = rem & 127;
    int hh = hhc << 1, ww = wwc << 1;

    const float* p  = pix + (size_t)b * 3u * 65536u;
    float*       Yp = Y   + (size_t)b * 65536u;

    float cbacc = 0.0f, cracc = 0.0f;
#pragma unroll
    for (int dw = 0; dw < 2; ++dw) {
        size_t off = (size_t)(ww + dw) * 256u + (size_t)hh;
        float2 R = *reinterpret_cast<const float2*>(p + 0u * 65536u + off);
        float2 G = *reinterpret_cast<const float2*>(p + 1u * 65536u + off);
        float2 B = *reinterpret_cast<const float2*>(p + 2u * 65536u + off);
        float2 yv;
        yv.x = 0.299f * R.x + 0.587f * G.x + 0.114f * B.x;
        yv.y = 0.299f * R.y + 0.587f * G.y + 0.114f * B.y;
        cbacc += (-0.168736f * R.x - 0.331264f * G.x + 0.5f * B.x)
               + (-0.168736f * R.y - 0.331264f * G.y + 0.5f * B.y);
        cracc += ( 0.5f * R.x - 0.418688f * G.x - 0.081312f * B.x)
               + ( 0.5f * R.y - 0.418688f * G.y - 0.081312f * B.y);
        *reinterpret_cast<float2*>(Yp + off) = yv;
    }
    size_t cidx = (size_t)b * 16384u + (size_t)wwc * 128u + (size_t)hhc;
    Cb[cidx] = 0.25f * cbacc + 128.0f;
    Cr[cidx] = 0.25f * cracc + 128.0f;
}

// ---------------------------------------------------------------------------
// FP32 WMMA helpers: 16x16x16 GEMM as 4 chained V_WMMA_F32_16X16X4_F32.
// A fragment layout (ISA 7.12.2, 32-bit A 16x4): lanes 0-15 hold K=0,1 in
// VGPR0/1; lanes 16-31 hold K=2,3.  B fragment assumed analogous to C/D
// pattern: VGPR r holds rows K=r (lanes 0-15) and K=r+2 (lanes 16-31).
// ---------------------------------------------------------------------------
__device__ __forceinline__ v8f gemm16(const v2f frag[4], const float* X, int lane)
{
    const int half = lane >> 4;
    const int lm   = lane & 15;
    v8f acc = {0.f, 0.f, 0.f, 0.f, 0.f, 0.f, 0.f, 0.f};
#pragma unroll
    for (int j = 0; j < 4; ++j) {
        int kb = 4 * j + 2 * half;
        v2f bf;
        bf.x = X[(kb + 0) * 17 + lm];
        bf.y = X[(kb + 1) * 17 + lm];
        acc = __builtin_amdgcn_wmma_f32_16x16x4_f32(
                  false, frag[j], false, bf, (short)0, acc, false, false);
    }
    return acc;
}

// ---------------------------------------------------------------------------
// Stage 2: in-place per-8x8-block DCT -> noisy quantize -> dequantize -> IDCT.
// One wave32 owns one 16x16 tile (2x2 blocks).  Block-diagonal DCT matrix
// M = diag(C,C), C[x][u] = cos((2x+1)u*pi/16):
//   F  = 0.25*AA .* (Mt P M)          (two WMMA chains, Xt trick in between)
//   E  = (F/tab + lam*0.5*rv)*tab .* AA
//   out= 0.25*(M E Mt) + 128          (two more WMMA chains)
// ---------------------------------------------------------------------------
__global__ void __launch_bounds__(128)
k_dct(float* plane, const float* __restrict__ rv, const float* __restrict__ eps,
      int Wn, int Hn, int isY)
{
    __shared__ float lds[4][16 * 17];
    const int lane = threadIdx.x & 31;
    const int wv   = threadIdx.x >> 5;
    const int half = lane >> 4;
    const int lm   = lane & 15;
    float* X = lds[wv];

    const int ntx = Hn >> 4;            // tiles along hh
    const int nty = Wn >> 4;            // tiles along ww
    const int tpi = ntx * nty;
    int gt  = blockIdx.x * 4 + wv;
    int b   = gt / tpi;
    int t   = gt - b * tpi;
    int thh = t % ntx;
    int tww = t / ntx;
    int hh0 = thh << 4, ww0 = tww << 4;

    // Constant A fragments: fMT = (Mt) chunked, fM = (M) chunked.
    v2f fMT[4], fM[4];
#pragma unroll
    for (int j = 0; j < 4; ++j) {
        int k0 = 4 * j + 2 * half;
#pragma unroll
        for (int e = 0; e < 2; ++e) {
            int  k  = k0 + e;
            bool on = (k >> 3) == (lm >> 3);      // block-diagonal
            float mt = on ? __cosf((float)((2 * (k & 7) + 1) * (lm & 7)) * PI_16) : 0.f;
            float mm = on ? __cosf((float)((2 * (lm & 7) + 1) * (k & 7)) * PI_16) : 0.f;
            if (e == 0) { fMT[j].x = mt; fM[j].x = mm; }
            else        { fMT[j].y = mt; fM[j].y = mm; }
        }
    }

    const float lam5 = 0.5f * eps[b];
    float* pp = plane + (size_t)b * (size_t)(Wn * Hn);

    // Load tile: X[x=hh_local][y=ww_local] = plane - 128   (row pitch 17)
#pragma unroll
    for (int i = 0; i < 8; ++i) {
        int idx = i * 32 + lane;
        int row = idx & 15, col = idx >> 4;
        X[row * 17 + col] = pp[(size_t)(ww0 + col) * Hn + (hh0 + row)] - 128.0f;
    }
    __syncthreads();

    v8f acc = gemm16(fMT, X, lane);                 // T1 = Mt * P
    __syncthreads();
#pragma unroll
    for (int r = 0; r < 8; ++r)                     // LDS <- T1^T
        X[lm * 17 + (r + 8 * half)] = acc[r];
    __syncthreads();
    acc = gemm16(fMT, X, lane);                     // acc = Mt*T1^T = F^T (raw)
    __syncthreads();

    // Elementwise: lane lm holds freq row u_t = lm, v_t = r + 8*half.
    {
        const float* tab = isY ? YTAB : CTAB;
        int u  = lm & 7;
        int bx = lm >> 3;                           // hh sub-block
        int by = half;                              // ww sub-block
        int nblkw = Wn >> 3;
        int n     = ((thh << 1) + bx) * nblkw + ((tww << 1) + by);
        int nblk  = (Hn >> 3) * nblkw;
        const float* rvp = rv + ((size_t)b * nblk + (size_t)n) * 64u + (size_t)u * 8u;
        float aau = (u == 0) ? RSQ2 : 1.0f;
#pragma unroll
        for (int r = 0; r < 8; ++r) {               // v = r
            float aav = aau * ((r == 0) ? RSQ2 : 1.0f);
            float tv  = tab[u * 8 + r];
            float f   = 0.25f * aav * acc[r];
            float q   = f / tv + lam5 * rvp[r];     // noisy rounding
            float e   = q * tv * aav;               // dequant * AA
            X[lm * 17 + (r + 8 * half)] = e;        // LDS <- E row-major [u][v]
        }
    }
    __syncthreads();

    acc = gemm16(fM, X, lane);                      // T2 = M * E
    __syncthreads();
#pragma unroll
    for (int r = 0; r < 8; ++r)                     // LDS <- T2^T
        X[lm * 17 + (r + 8 * half)] = acc[r];
    __syncthreads();
    acc = gemm16(fM, X, lane);                      // acc = M*T2^T = out^T (raw)

    // Store: value is out[x = lm][y = r + 8*half]; lanes 0-15 contiguous in hh.
#pragma unroll
    for (int r = 0; r < 8; ++r) {
        int y = r + 8 * half;
        pp[(size_t)(ww0 + y) * Hn + (hh0 + lm)] = 0.25f * acc[r] + 128.0f;
    }
}

// ---------------------------------------------------------------------------
// Stage 3: nearest 2x chroma upsample + YCbCr -> RGB + clip, coalesced float2.
// ---------------------------------------------------------------------------
__global__ void k_ycc2rgb(const float* __restrict__ Y, const float* __restrict__ Cb,
                          const float* __restrict__ Cr, float* __restrict__ out)
{
    int g   = blockIdx.x * blockDim.x + threadIdx.x;
    int b   = g >> 14;
    int rem = g & 16383;
    int wwc = rem >> 7;
    int hhc = rem & 127;
    int hh = hhc << 1, ww = wwc << 1;

    size_t cidx = (size_t)b * 16384u + (size_t)wwc * 128u + (size_t)hhc;
    float cb = Cb[cidx] - 128.0f;
    float cr = Cr[cidx] - 128.0f;

    const float* Yp = Y   + (size_t)b * 65536u;
    float*       op = out + (size_t)b * 3u * 65536u;

#pragma unroll
    for (int dw = 0; dw < 2; ++dw) {
        size_t off = (size_t)(ww + dw) * 256u + (size_t)hh;
        float2 yv = *reinterpret_cast<const float2*>(Yp + off);
        float2 R, G, B;
        R.x = yv.x + 1.402f * cr;
        R.y = yv.y + 1.402f * cr;
        G.x = yv.x - 0.344136f * cb - 0.714136f * cr;
        G.y = yv.y - 0.344136f * cb - 0.714136f * cr;
        B.x = yv.x + 1.772f * cb;
        B.y = yv.y + 1.772f * cb;
        R.x = fminf(fmaxf(R.x, 0.f), 255.f);  R.y = fminf(fmaxf(R.y, 0.f), 255.f);
        G.x = fminf(fmaxf(G.x, 0.f), 255.f);  G.y = fminf(fmaxf(G.y, 0.f), 255.f);
        B.x = fminf(fmaxf(B.x, 0.f), 255.f);  B.y = fminf(fmaxf(B.y, 0.f), 255.f);
        *reinterpret_cast<float2*>(op + 0u * 65536u + off) = R;
        *reinterpret_cast<float2*>(op + 1u * 65536u + off) = G;
        *reinterpret_cast<float2*>(op + 2u * 65536u + off) = B;
    }
}

// ---------------------------------------------------------------------------
extern "C" void kernel_launch(void* const* d_in, const int* in_sizes, int n_in,
                              void* d_out, int out_size, void* d_ws, size_t ws_size,
                              hipStream_t stream)
{
    (void)in_sizes; (void)n_in; (void)out_size; (void)ws_size;
    const float* pix   = (const float*)d_in[0];
    const float* rv_y  = (const float*)d_in[1];
    const float* rv_cb = (const float*)d_in[2];
    const float* rv_cr = (const float*)d_in[3];
    const float* eps   = (const float*)d_in[4];
    float* out = (float*)d_out;

    float* ws = (float*)d_ws;
    float* Y  = ws;                                   // 64 * 256*256 floats
    float* Cb = Y  + (size_t)64 * 65536u;             // 64 * 128*128 floats
    float* Cr = Cb + (size_t)64 * 16384u;             // 64 * 128*128 floats

    // 64 images * 128*128 quads = 1,048,576 threads
    k_rgb2ycc<<<4096, 256, 0, stream>>>(pix, Y, Cb, Cr);

    // Y: 64 * 16*16 tiles = 16384 tiles, 4 waves (tiles) per block
    k_dct<<<4096, 128, 0, stream>>>(Y,  rv_y,  eps, 256, 256, 1);
    // Cb/Cr: 64 * 8*8 tiles = 4096 tiles each
    k_dct<<<1024, 128, 0, stream>>>(Cb, rv_cb, eps, 128, 128, 0);
    k_dct<<<1024, 128, 0, stream>>>(Cr, rv_cr, eps, 128, 128, 0);

    k_ycc2rgb<<<4096, 256, 0, stream>>>(Y, Cb, Cr, out);
}